// FIRE_64527588655149
// MI455X (gfx1250) — compile-verified
//
#include <hip/hip_runtime.h>
#include <stdint.h>

typedef float v4f __attribute__((ext_vector_type(4)));
typedef unsigned int u32x4 __attribute__((ext_vector_type(4)));
typedef int          i32x4 __attribute__((ext_vector_type(4)));
typedef int          i32x8 __attribute__((ext_vector_type(8)));

#define T_  32
#define C_  3
#define H_  720
#define W_  1280
#define N_  30                       // T-2 output frames
#define HW_ (H_ * W_)                // 921600
#define W4_ (W_ / 4)                 // 320 float4 groups per row == blockDim.x

#if __has_builtin(__builtin_amdgcn_tensor_load_to_lds)
#define HAVE_TDM 1
#else
#define HAVE_TDM 0
#endif

// TDM-as-prefetcher: DMA one 1280-float tensor row into an LDS scratch region
// that is never read. The payload lands in L2 on the way, so the block's
// immediately-following direct loads of the same row hit L2. A wrong D# can
// only waste a 5KB DMA - it cannot corrupt results.
// D# per CDNA5 ISA ch.8: group0 = {count=1 | type=2, lds_addr, gaddr[56:0]},
// group1 = {data_size=4B, tensor_dim0=1280, tensor_dim1=1, tile=1280x1,
//           dim0_stride=1280}, groups 2/3 zero (<=2D tensor).
__device__ __forceinline__ void tdm_prefetch_row(const float* gsrc, unsigned lds_byte_off) {
#if HAVE_TDM
    unsigned long long ga = (unsigned long long)(uintptr_t)gsrc;
    u32x4 g0 = { 1u,                                                   // count=1, user mode
                 lds_byte_off,                                         // lds_addr
                 (unsigned)(ga & 0xffffffffu),                         // global_addr[31:0]
                 (unsigned)((ga >> 32) & 0x01ffffffu) | (2u << 30) };  // addr[56:32] | type=2
    i32x8 g1 = { (int)(2u << 16),        // wg_mask=0, data_size=2 (4B), no pad/iter/barrier
                 (int)((unsigned)W_ << 16),   // tensor_dim0[15:0] = 1280
                 (int)(1u << 16),             // tensor_dim0 hi = 0 | tensor_dim1 = 1
                 (int)((unsigned)W_ << 16),   // tensor_dim1 hi = 0 | tile_dim0 = 1280
                 1,                           // tile_dim1 = 1, tile_dim2 = 0
                 W_,                          // tensor_dim0_stride = 1280
                 0, 0 };                      // stride hi, tensor_dim1_stride = 0
    i32x4 g2 = { 0, 0, 0, 0 };
    i32x4 g3 = { 0, 0, 0, 0 };
#if __clang_major__ >= 23
    i32x8 g4 = { 0, 0, 0, 0, 0, 0, 0, 0 };
    __builtin_amdgcn_tensor_load_to_lds(g0, g1, g2, g3, g4, 0);
#else
    __builtin_amdgcn_tensor_load_to_lds(g0, g1, g2, g3, 0);
#endif
#else
    (void)gsrc; (void)lds_byte_off;
#endif
}

// One block = one (n, h) row: 320 threads x 4 w-pixels = 1280 = W.
// n, h come from blockIdx -> all base addresses are SGPR (saddr-form loads);
// only threadIdx.x*4 is a vector offset. 320 threads = 10 wave32s.
__global__ __launch_bounds__(W4_) void FIRE_flowdiff_kernel(
    const float* __restrict__ x,      // (32, 3, 720, 1280)
    const float* __restrict__ flow,   // (30, 2, 720, 1280)
    float* __restrict__ out)          // (30, 6, 720, 1280)
{
    const int h  = blockIdx.x;        // 0..719
    const int n  = blockIdx.y;        // 0..29
    const int m  = (N_ - 1) - n;      // bwd flow frame index
    const int w0 = threadIdx.x * 4;   // 0..1276

    const size_t rowbase = (size_t)h * W_;           // scalar
    const float* pf0 = flow + (size_t)(2 * n    ) * HW_ + rowbase;
    const float* pf1 = flow + (size_t)(2 * n + 1) * HW_ + rowbase;
    const float* pb0 = flow + (size_t)(2 * m    ) * HW_ + rowbase;
    const float* pb1 = flow + (size_t)(2 * m + 1) * HW_ + rowbase;

#if HAVE_TDM
    __shared__ float tdm_scratch[W_];                // write-only DMA target (5 KB)
    if (threadIdx.x == 0) {                          // wave 0 only; others EXECZ-skip
        unsigned lds0 = (unsigned)(uintptr_t)(void*)tdm_scratch;
        tdm_prefetch_row(pf0, lds0);
        tdm_prefetch_row(pf1, lds0);
        tdm_prefetch_row(pb0, lds0);
        tdm_prefetch_row(pb1, lds0);
        __builtin_amdgcn_s_wait_tensorcnt(0);
    }
#endif

    // Flow is read exactly twice across the whole kernel (fwd at n, bwd at 29-n,
    // far apart in time) -> non-temporal, keep L2 for the reused x frames.
    v4f fx = __builtin_nontemporal_load((const v4f*)(pf0 + w0));
    v4f fy = __builtin_nontemporal_load((const v4f*)(pf1 + w0));
    v4f bx = __builtin_nontemporal_load((const v4f*)(pb0 + w0));
    v4f by = __builtin_nontemporal_load((const v4f*)(pb1 + w0));

    const float* frameC = x + (size_t)(n + 1) * C_ * HW_;   // center (xc)
    const float* frameF = x + (size_t)(n + 2) * C_ * HW_;   // fwd gather source
    const float* frameB = x + (size_t)(n    ) * C_ * HW_;   // bwd gather source

    // Pre-touch center of the gather window (flow ~ N(0,4) => +/-12 px)
    __builtin_prefetch(frameF + rowbase + w0, 0, 3);
    __builtin_prefetch(frameB + rowbase + w0, 0, 3);

    // Index math once, shared by all 3 channels. jnp.round == RNE -> rintf.
    int offF[4], offB[4];
#pragma unroll
    for (int j = 0; j < 4; ++j) {
        float ixf = fminf(fmaxf(rintf((float)h        + fx[j]), 0.f), (float)(H_ - 1));
        float iyf = fminf(fmaxf(rintf((float)(w0 + j) + fy[j]), 0.f), (float)(W_ - 1));
        float ixb = fminf(fmaxf(rintf((float)h        + bx[j]), 0.f), (float)(H_ - 1));
        float iyb = fminf(fmaxf(rintf((float)(w0 + j) + by[j]), 0.f), (float)(W_ - 1));
        offF[j] = (int)ixf * W_ + (int)iyf;
        offB[j] = (int)ixb * W_ + (int)iyb;
    }

    float* outF = out + (size_t)n * 6 * HW_ + rowbase + w0;  // ch 0..2: xc - x_fwd
    float* outB = outF + (size_t)3 * HW_;                    // ch 3..5: xc - x_bwd

#pragma unroll
    for (int c = 0; c < C_; ++c) {
        const float* fc = frameC + (size_t)c * HW_;
        const float* ff = frameF + (size_t)c * HW_;
        const float* fb = frameB + (size_t)c * HW_;

        v4f xc = *(const v4f*)(fc + rowbase + w0);           // temporal (reused frames)

        v4f gF, gB;
#pragma unroll
        for (int j = 0; j < 4; ++j) {
            gF[j] = ff[offF[j]];
            gB[j] = fb[offB[j]];
        }

        __builtin_nontemporal_store(xc - gF, (v4f*)(outF + (size_t)c * HW_));
        __builtin_nontemporal_store(xc - gB, (v4f*)(outB + (size_t)c * HW_));
    }
}

extern "C" void kernel_launch(void* const* d_in, const int* in_sizes, int n_in,
                              void* d_out, int out_size, void* d_ws, size_t ws_size,
                              hipStream_t stream) {
    (void)in_sizes; (void)n_in; (void)out_size; (void)d_ws; (void)ws_size;
    const float* x    = (const float*)d_in[0];
    const float* flow = (const float*)d_in[1];
    float*       out  = (float*)d_out;

    // One block per (h, n) row; 320 threads cover the full W=1280 row as float4s.
    FIRE_flowdiff_kernel<<<dim3(H_, N_), dim3(W4_), 0, stream>>>(x, flow, out);
}